// Net_25907242729900
// MI455X (gfx1250) — compile-verified
//
#include <hip/hip_runtime.h>
#include <hip/hip_bf16.h>
#include <math.h>

typedef __attribute__((ext_vector_type(2))) float v2f;
typedef __attribute__((ext_vector_type(8))) float v8f;

// ---------------- utility fills ----------------
__global__ void k_fill(float* __restrict__ p, long long n, float v) {
  long long i = (long long)blockIdx.x * blockDim.x + threadIdx.x;
  if (i < n) p[i] = v;
}

// ---------------- degree / norm ----------------
__global__ void k_degree(const long long* __restrict__ dst, float* __restrict__ deg, int E) {
  int e = blockIdx.x * blockDim.x + threadIdx.x;
  if (e < E) atomicAdd(&deg[(int)dst[e]], 1.0f);
}

__global__ void k_rsqrt_ip(float* __restrict__ d, int N) {
  int i = blockIdx.x * blockDim.x + threadIdx.x;
  if (i < N) d[i] = rsqrtf(d[i]);   // deg >= 1 always (self-loop)
}

// ---------------- GEMM1: h1 = x @ W1   (N x 512) @ (512 x 16) ----------------
// One wave computes one 16x16 output tile via V_WMMA_F32_16X16X4_F32, K-loop 128 steps.
__global__ void __launch_bounds__(256)
k_gemm1(const float* __restrict__ x, const float* __restrict__ W,
        float* __restrict__ h, int Ntiles) {
  __shared__ float sW[512 * 16];                     // 32 KB: whole W1
  for (int i = threadIdx.x; i < 512 * 16; i += 256) sW[i] = W[i];
  __syncthreads();

  int wave = threadIdx.x >> 5;
  int lane = threadIdx.x & 31;
  int tile = blockIdx.x * 8 + wave;                  // wave-uniform
  if (tile >= Ntiles) return;

  int m  = lane & 15;                                // A row / B col
  int kb = (lane >> 4) << 1;                         // 0 or 2
  const float* xr = x + (size_t)(tile * 16 + m) * 512 + kb;

  v8f c = {};
#pragma unroll 4
  for (int k0 = 0; k0 < 512; k0 += 4) {
    v2f a, b;
    a.x = xr[k0];
    a.y = xr[k0 + 1];
    b.x = sW[(k0 + kb) * 16 + m];
    b.y = sW[(k0 + kb + 1) * 16 + m];
    c = __builtin_amdgcn_wmma_f32_16x16x4_f32(false, a, false, b, (short)0, c, false, false);
  }

  int rbase = tile * 16 + ((lane >> 4) << 3);        // M = v + 8*(lane>>4)
#pragma unroll
  for (int v = 0; v < 8; ++v)
    h[(size_t)(rbase + v) * 16 + m] = c[v];
}

// ---------------- edge aggregation layer 1 (H=16) ----------------
__global__ void k_agg1(const long long* __restrict__ src, const long long* __restrict__ dst,
                       const float* __restrict__ h1, const float* __restrict__ dinv,
                       float* __restrict__ agg, unsigned total) {
  unsigned t = blockIdx.x * blockDim.x + threadIdx.x;
  if (t >= total) return;
  unsigned e = t >> 4;
  unsigned j = t & 15u;
  int s = (int)src[e];
  int d = (int)dst[e];
  float norm = dinv[s] * dinv[d];
  atomicAdd(&agg[(size_t)d * 16 + j], h1[(size_t)s * 16 + j] * norm);
}

// ---------------- self-loop + bias + ELU (in place into agg1 -> x1) ----------------
__global__ void k_fin1(const float* __restrict__ h1, const float* __restrict__ b1,
                       const float* __restrict__ dinv, float* __restrict__ agg, unsigned total) {
  unsigned t = blockIdx.x * blockDim.x + threadIdx.x;
  if (t >= total) return;
  unsigned n = t >> 4;
  unsigned j = t & 15u;
  float di = dinv[n];
  float v = agg[t] + h1[t] * di * di + b1[j];
  agg[t] = (v > 0.0f) ? v : expm1f(v);               // ELU, alpha=1
}

// ---------------- GEMM2: h2 = x1 @ W2   (N x 16) @ (16 x 40), 40 cols padded to 48 ----------------
__global__ void __launch_bounds__(256)
k_gemm2(const float* __restrict__ x1, const float* __restrict__ W2,
        float* __restrict__ h2, int Nrt) {
  __shared__ float sW[16 * 48];                      // W2 zero-padded to 16x48
  for (int i = threadIdx.x; i < 16 * 48; i += 256) {
    int r = i / 48, cc = i % 48;
    sW[i] = (cc < 40) ? W2[r * 40 + cc] : 0.0f;
  }
  __syncthreads();

  int wave = threadIdx.x >> 5;
  int lane = threadIdx.x & 31;
  int tile = blockIdx.x * 8 + wave;                  // wave-uniform
  if (tile >= Nrt * 3) return;
  int rt = tile / 3;
  int ct = tile % 3;

  int m  = lane & 15;
  int kb = (lane >> 4) << 1;
  const float* xr = x1 + (size_t)(rt * 16 + m) * 16 + kb;

  v8f c = {};
#pragma unroll
  for (int k0 = 0; k0 < 16; k0 += 4) {
    v2f a, b;
    a.x = xr[k0];
    a.y = xr[k0 + 1];
    b.x = sW[(k0 + kb) * 48 + ct * 16 + m];
    b.y = sW[(k0 + kb + 1) * 48 + ct * 16 + m];
    c = __builtin_amdgcn_wmma_f32_16x16x4_f32(false, a, false, b, (short)0, c, false, false);
  }

  int col = ct * 16 + m;                             // guard AFTER wmma (EXEC all-ones inside)
  if (col < 40) {
    int rbase = rt * 16 + ((lane >> 4) << 3);
#pragma unroll
    for (int v = 0; v < 8; ++v)
      h2[(size_t)(rbase + v) * 40 + col] = c[v];
  }
}

// ---------------- edge aggregation layer 2 (C=40) ----------------
__global__ void k_agg2(const long long* __restrict__ src, const long long* __restrict__ dst,
                       const float* __restrict__ h2, const float* __restrict__ dinv,
                       float* __restrict__ agg, unsigned total) {
  unsigned t = blockIdx.x * blockDim.x + threadIdx.x;
  if (t >= total) return;
  unsigned e = t / 40u;
  unsigned cch = t - e * 40u;
  int s = (int)src[e];
  int d = (int)dst[e];
  float norm = dinv[s] * dinv[d];
  atomicAdd(&agg[(size_t)d * 40 + cch], h2[(size_t)s * 40 + cch] * norm);
}

// ---------------- self-loop + bias + log_softmax -> d_out ----------------
__global__ void k_fin2(const float* __restrict__ h2, const float* __restrict__ b2,
                       const float* __restrict__ dinv, const float* __restrict__ agg,
                       float* __restrict__ out, int N) {
  int n = blockIdx.x * blockDim.x + threadIdx.x;
  if (n >= N) return;
  float di = dinv[n];
  float di2 = di * di;
  float l[40];
  float mx = -INFINITY;
#pragma unroll
  for (int c = 0; c < 40; ++c) {
    float v = agg[(size_t)n * 40 + c] + h2[(size_t)n * 40 + c] * di2 + b2[c];
    l[c] = v;
    mx = fmaxf(mx, v);
  }
  float s = 0.0f;
#pragma unroll
  for (int c = 0; c < 40; ++c) s += expf(l[c] - mx);
  float lse = mx + logf(s);
#pragma unroll
  for (int c = 0; c < 40; ++c) out[(size_t)n * 40 + c] = l[c] - lse;
}

extern "C" void kernel_launch(void* const* d_in, const int* in_sizes, int n_in,
                              void* d_out, int out_size, void* d_ws, size_t ws_size,
                              hipStream_t stream) {
  const float*     x   = (const float*)d_in[0];
  const float*     W1  = (const float*)d_in[1];
  const float*     b1  = (const float*)d_in[2];
  const float*     W2  = (const float*)d_in[3];
  const float*     b2  = (const float*)d_in[4];
  const long long* ei  = (const long long*)d_in[5];

  const int H = in_sizes[2];               // 16
  const int F = in_sizes[1] / H;           // 512
  const int N = in_sizes[0] / F;           // 100000
  const int E = in_sizes[5] / 2;           // 3200000
  (void)n_in; (void)out_size; (void)ws_size; (void)H; (void)F;

  const long long* src = ei;
  const long long* dst = ei + E;

  // workspace layout (floats): dinv | h1 | agg1(->x1) | h2 | agg2   ~45 MB
  float* ws   = (float*)d_ws;
  float* dinv = ws;
  float* h1   = dinv + N;
  float* agg1 = h1 + (size_t)N * 16;
  float* h2   = agg1 + (size_t)N * 16;
  float* agg2 = h2 + (size_t)N * 40;

  // init (re-done every call: graph-replay safe, no stale state)
  k_fill<<<(N + 255) / 256, 256, 0, stream>>>(dinv, N, 1.0f);              // self-loop degree
  k_fill<<<(int)(((long long)N * 16 + 255) / 256), 256, 0, stream>>>(agg1, (long long)N * 16, 0.0f);
  k_fill<<<(int)(((long long)N * 40 + 255) / 256), 256, 0, stream>>>(agg2, (long long)N * 40, 0.0f);

  k_degree<<<(E + 255) / 256, 256, 0, stream>>>(dst, dinv, E);
  k_rsqrt_ip<<<(N + 255) / 256, 256, 0, stream>>>(dinv, N);

  const int nt = N / 16;                   // 6250, exact
  k_gemm1<<<(nt + 7) / 8, 256, 0, stream>>>(x, W1, h1, nt);

  unsigned tot1 = (unsigned)E * 16u;
  k_agg1<<<(tot1 + 255) / 256, 256, 0, stream>>>(src, dst, h1, dinv, agg1, tot1);

  unsigned totf1 = (unsigned)N * 16u;
  k_fin1<<<(totf1 + 255) / 256, 256, 0, stream>>>(h1, b1, dinv, agg1, totf1);

  k_gemm2<<<(nt * 3 + 7) / 8, 256, 0, stream>>>(agg1, W2, h2, nt);

  unsigned tot2 = (unsigned)E * 40u;
  k_agg2<<<(tot2 + 255) / 256, 256, 0, stream>>>(src, dst, h2, dinv, agg2, tot2);

  k_fin2<<<(N + 255) / 256, 256, 0, stream>>>(h2, b2, dinv, agg2, (float*)d_out, N);
}